// FastSpeech2Decoder_85194971283824
// MI455X (gfx1250) — compile-verified
//
#include <hip/hip_runtime.h>
#include <stdint.h>

// Shapes fixed by the reference.
#define T_FRAMES 8192
#define T_TOK    1024
#define NTHR     256
#define SEG      32                       // frames per thread
#define PAD32(i) ((i) + ((i) >> 5))       // +1 dword pad per 32 -> bank-conflict-free stride 33

// 1 = use Tensor Data Mover (one tensor_load_to_lds per block, HW-padded layout)
// 0 = fall back to per-lane global_load_async_to_lds_b32
#ifndef USE_TDM
#define USE_TDM 1
#endif

typedef uint32_t u32x4 __attribute__((ext_vector_type(4)));
typedef uint32_t u32x8 __attribute__((ext_vector_type(8)));

// ---- CDNA5 async / tensor ops (gfx1250) -----------------------------------
#if defined(__HIP_DEVICE_COMPILE__)

#if __has_builtin(__builtin_amdgcn_s_wait_asynccnt)
#define WAIT_ASYNC0() __builtin_amdgcn_s_wait_asynccnt(0)
#else
#define WAIT_ASYNC0() asm volatile("s_wait_asynccnt 0" ::: "memory")
#endif

#if __has_builtin(__builtin_amdgcn_s_wait_tensorcnt)
#define WAIT_TENSOR0() __builtin_amdgcn_s_wait_tensorcnt(0)
#else
#define WAIT_TENSOR0() asm volatile("s_wait_tensorcnt 0" ::: "memory")
#endif

__device__ __forceinline__ void async_g2l_b32(uint32_t lds_byte, uint64_t gbase, uint32_t gbyte) {
  asm volatile("global_load_async_to_lds_b32 %0, %1, %2"
               :: "v"(lds_byte), "v"(gbyte), "s"(gbase)
               : "memory");
}

__device__ __forceinline__ void async_g2l_b128(uint32_t lds_byte, uint64_t gbase, uint32_t gbyte) {
  asm volatile("global_load_async_to_lds_b128 %0, %1, %2"
               :: "v"(lds_byte), "v"(gbyte), "s"(gbase)
               : "memory");
}

// TDM: copy one 8192-dword row Global->LDS, inserting 1 pad dword per 32 dwords
// (pad_interval field 4 => 32 dwords; pad_amount field 0 => 1 dword), i.e. the
// PAD32() layout, produced by the DMA engine instead of the wave.
__device__ __forceinline__ void tdm_load_row_padded(uint32_t lds_byte, uint64_t gaddr) {
  u32x4 g0;
  g0.x = 1u;                                            // count=1, user mode
  g0.y = lds_byte;                                      // lds_addr
  g0.z = (uint32_t)gaddr;                               // global_addr[31:0]
  g0.w = ((uint32_t)(gaddr >> 32) & 0x01FFFFFFu)        // global_addr[56:32]
         | (2u << 30);                                  // type=2 ("image")
  u32x8 g1;
  g1[0] = (2u << 16)            // data_size = 4 bytes
        | (1u << 20)            // pad_enable
        | (4u << 22)            // pad_interval: 1<<4 * 8B = 128B = 32 dwords
        | (0u << 25);           // pad_amount: 1 dword
  g1[1] = (uint32_t)T_FRAMES << 16;   // tensor_dim0[15:0] = 8192 (bits 63:48)
  g1[2] = 1u << 16;                    // tensor_dim0[31:16]=0 | tensor_dim1[15:0]=1
  g1[3] = (uint32_t)T_FRAMES << 16;   // tensor_dim1[31:16]=0 | tile_dim0 = 8192
  g1[4] = 1u;                          // tile_dim1 = 1, tile_dim2 = 0
  g1[5] = (uint32_t)T_FRAMES;          // tensor_dim0_stride[31:0] = 8192
  g1[6] = 0u;                          // stride hi / dim1_stride lo
  g1[7] = 0u;
  asm volatile("tensor_load_to_lds %0, %1" :: "s"(g0), "s"(g1) : "memory");
}

#else
#define WAIT_ASYNC0()
#define WAIT_TENSOR0()
__device__ __forceinline__ void async_g2l_b32(uint32_t, uint64_t, uint32_t) {}
__device__ __forceinline__ void async_g2l_b128(uint32_t, uint64_t, uint32_t) {}
__device__ __forceinline__ void tdm_load_row_padded(uint32_t, uint64_t) {}
#endif

// ---- 256-wide exclusive scan (Hillis-Steele in LDS) -----------------------
__device__ __forceinline__ float block_exscan(float v, float* sa, float* sb, int t) {
  sa[t] = v;
  __syncthreads();
  float* src = sa;
  float* dst = sb;
#pragma unroll
  for (int d = 1; d < NTHR; d <<= 1) {
    float x = src[t];
    if (t >= d) x += src[t - d];
    dst[t] = x;
    __syncthreads();
    float* tmp = src; src = dst; dst = tmp;
  }
  float ex = (t > 0) ? src[t - 1] : 0.f;   // exclusive prefix
  __syncthreads();                          // buffers safe for reuse
  return ex;
}

__device__ __forceinline__ float seg_avg(float s, float e, const float* cfs) {
  float d = e - s;                          // integer-valued in f32
  if (!(d > 0.f)) return 0.f;
  int si = (int)s;
  int ei = (int)e;
  return (cfs[PAD32(ei)] - cfs[PAD32(si)]) / d;
}

__global__ void __launch_bounds__(NTHR) fs2_dur_avg_kernel(
    const float* __restrict__ frame_scalar,   // [B, T_FRAMES]
    const float* __restrict__ duration,       // [B, T_TOK]
    const int*   __restrict__ frame_len,      // [B]
    const int*   __restrict__ dur_len,        // [B]
    float*       __restrict__ out_tok,        // [B, T_TOK]
    int*         __restrict__ out_len)        // [B]
{
  // s_cfs doubles as: (a) padded raw frame row, (b) padded exclusive prefix sums.
  __shared__ __align__(16) float  s_cfs[PAD32(T_FRAMES) + 1];   // 8449 dwords
  __shared__ float4 s_dur4[NTHR];                               // 1024 floats
  __shared__ float  s_sa[NTHR];
  __shared__ float  s_sb[NTHR];

  const int b = blockIdx.x;
  const int t = threadIdx.x;

  const uint64_t frow = (uint64_t)(uintptr_t)(frame_scalar + (size_t)b * T_FRAMES);
  const uint64_t drow = (uint64_t)(uintptr_t)(duration     + (size_t)b * T_TOK);

  // ---- stage rows into LDS ----
  const uint32_t cfs_base = (uint32_t)(uintptr_t)(void*)&s_cfs[0];
#if USE_TDM
  if (t < 32) {                       // TDM ignores EXEC: exactly one wave issues it
    tdm_load_row_padded(cfs_base, frow);
  }
#else
#pragma unroll
  for (int p = 0; p < T_FRAMES / NTHR; ++p) {
    uint32_t i = (uint32_t)(p * NTHR + t);
    async_g2l_b32(cfs_base + 4u * PAD32(i), frow, 4u * i);
  }
#endif
  {
    const uint32_t dur_base = (uint32_t)(uintptr_t)(void*)&s_dur4[0];
    async_g2l_b128(dur_base + 16u * (uint32_t)t, drow, 16u * (uint32_t)t);
  }

  const int flen = frame_len[b];
  const int dlen = dur_len[b];

  WAIT_ASYNC0();
  WAIT_TENSOR0();                     // no-op for waves with TENSORcnt==0
  __syncthreads();

  // ---- per-thread inclusive prefix over its 32 contiguous frames ----
  float loc[SEG];
  {
    const float* rp = &s_cfs[33 * t];                  // stride 33: bank-conflict-free
    float run = 0.f;
#pragma unroll
    for (int k = 0; k < SEG; ++k) { run += rp[k]; loc[k] = run; }
  }
  const float off = block_exscan(loc[SEG - 1], s_sa, s_sb, t);

  // ---- write exclusive prefix sums, aliased over the raw buffer ----
  // Safe: every thread's raw reads completed before the scan's first barrier.
  if (t == 0) s_cfs[0] = 0.f;
#pragma unroll
  for (int k = 0; k < SEG; ++k) {
    int j = SEG * t + k + 1;                           // cfs index 1..8192
    s_cfs[PAD32(j)] = off + loc[k];
  }
  __syncthreads();

  // ---- token phase: 4 tokens per thread ----
  const float4 dv = s_dur4[t];
  const int j0 = 4 * t;
  // round-half-even (rintf) + clamp at 0 + mask beyond duration_len
  const float d0 = (j0 + 0 < dlen) ? fmaxf(rintf(dv.x), 0.f) : 0.f;
  const float d1 = (j0 + 1 < dlen) ? fmaxf(rintf(dv.y), 0.f) : 0.f;
  const float d2 = (j0 + 2 < dlen) ? fmaxf(rintf(dv.z), 0.f) : 0.f;
  const float d3 = (j0 + 3 < dlen) ? fmaxf(rintf(dv.w), 0.f) : 0.f;

  const float l1 = d0;
  const float l2 = l1 + d1;
  const float l3 = l2 + d2;
  const float l4 = l3 + d3;

  const float toff = block_exscan(l4, s_sa, s_sb, t);  // frames consumed before token 4t
  const float fl = (float)flen;

  const float s0 = fminf(toff, fl);
  const float e0 = fminf(toff + l1, fl);
  const float e1 = fminf(toff + l2, fl);
  const float e2 = fminf(toff + l3, fl);
  const float e3 = fminf(toff + l4, fl);

  float4 o;
  o.x = seg_avg(s0, e0, s_cfs);
  o.y = seg_avg(e0, e1, s_cfs);
  o.z = seg_avg(e1, e2, s_cfs);
  o.w = seg_avg(e2, e3, s_cfs);

  reinterpret_cast<float4*>(out_tok + (size_t)b * T_TOK)[t] = o;
  if (t == 0) out_len[b] = dlen;                       // tuple output #2 (int32 bits)
}

extern "C" void kernel_launch(void* const* d_in, const int* in_sizes, int n_in,
                              void* d_out, int out_size, void* d_ws, size_t ws_size,
                              hipStream_t stream) {
  const float* frame_scalar = (const float*)d_in[0];
  const float* duration     = (const float*)d_in[1];
  const int*   frame_len    = (const int*)d_in[2];
  const int*   dur_len      = (const int*)d_in[3];

  const int B = in_sizes[2];                 // 2048
  float* out_tok = (float*)d_out;            // [B, T_TOK]
  int*   out_len = (int*)(out_tok + (size_t)B * T_TOK);  // [B], raw int32 bits

  fs2_dur_avg_kernel<<<dim3(B), dim3(NTHR), 0, stream>>>(
      frame_scalar, duration, frame_len, dur_len, out_tok, out_len);
}